// StackLogicClassifier_69458211111038
// MI455X (gfx1250) — compile-verified
//
#include <hip/hip_runtime.h>

#define NS      16384   // samples
#define KN      4096    // neurons per layer
#define NLAYERS 4
#define S       4       // samples per workgroup
#define BLOCK   256     // 8 waves (wave32)

typedef float v2f __attribute__((ext_vector_type(2)));
typedef float v8f __attribute__((ext_vector_type(8)));

// ---------------------------------------------------------------------------
// Kernel 1: collapse softmax(w) into the 4 multilinear coefficients per neuron
// out layout: coef[layer*KN + k] = (c0, ca, cb, cab)
// ---------------------------------------------------------------------------
__global__ void difflogic_coeff_kernel(const float* __restrict__ w0,
                                       const float* __restrict__ ws,
                                       float4* __restrict__ coef) {
    int t = blockIdx.x * blockDim.x + threadIdx.x;
    if (t >= NLAYERS * KN) return;
    int layer = t / KN, k = t - layer * KN;
    const float* w = (layer == 0) ? (w0 + (size_t)k * 16)
                                  : (ws + ((size_t)(layer - 1) * KN + k) * 16);
    float p[16];
    float mx = -3.402823466e38f;
    #pragma unroll
    for (int i = 0; i < 16; ++i) { p[i] = w[i]; mx = fmaxf(mx, p[i]); }
    float sum = 0.f;
    #pragma unroll
    for (int i = 0; i < 16; ++i) { p[i] = expf(p[i] - mx); sum += p[i]; }
    float inv = 1.0f / sum;
    #pragma unroll
    for (int i = 0; i < 16; ++i) p[i] *= inv;

    float ca  = p[2] + p[3] + p[6] + p[7] - p[8] - p[9] - p[12] - p[13];
    float cb  = p[4] + p[5] + p[6] + p[7] - p[8] - p[9] - p[10] - p[11];
    float cab = p[1] - p[2] - p[4] - 2.0f * p[6] - p[7] + p[8]
              + 2.0f * p[9] + p[11] + p[13] - p[14];
    float c0  = p[8] + p[9] + p[10] + p[11] + p[12] + p[13] + p[14] + p[15];
    coef[t] = make_float4(c0, ca, cb, cab);
}

// ---------------------------------------------------------------------------
// Kernel 2: fully fused two-pass model evaluation, 4 samples per workgroup.
// Dynamic LDS: 2 * S * KN floats (double-buffered layer activations).
// Final class-sums computed with v_wmma_f32_16x16x4_f32 (B = ones).
// ---------------------------------------------------------------------------
__global__ void difflogic_model_kernel(const float* __restrict__ x,
                                       const float4* __restrict__ coef,
                                       const int* __restrict__ ia0,
                                       const int* __restrict__ ib0,
                                       const int* __restrict__ ia,
                                       const int* __restrict__ ib,
                                       float* __restrict__ out) {
    extern __shared__ float smem[];            // [2][S][KN]
    __shared__ float s_mx[S][4];
    __shared__ float s_iv[S][4];
    __shared__ float s_rt[S][2];
    __shared__ float s_part[8][16];            // per-wave WMMA row partials

    const int tid   = threadIdx.x;
    const int wave  = tid >> 5;
    const int lane  = tid & 31;
    const int sbase = blockIdx.x * S;

    // ---- initial binarization: mx = (x0, x1, -x0, -x1); iv = (mx > 0) ----
    if (tid < S * 2) {
        int s = tid >> 1, j = tid & 1;
        float xv = x[(size_t)(sbase + s) * 2 + j];
        s_mx[s][j]     = xv;
        s_mx[s][2 + j] = -xv;
        s_iv[s][j]     = (xv > 0.0f)    ? 1.0f : 0.0f;
        s_iv[s][2 + j] = ((-xv) > 0.0f) ? 1.0f : 0.0f;
    }
    __syncthreads();

    for (int pass = 0; pass < 2; ++pass) {
        // ---------------- layer 0 (gather from 4-wide iv) ----------------
        {
            float* h0 = smem;   // buffer 0
            for (int k = tid; k < KN; k += BLOCK) {
                int    a_i = ia0[k];
                int    b_i = ib0[k];
                float4 c   = coef[k];
                #pragma unroll
                for (int s = 0; s < S; ++s) {
                    float a = s_iv[s][a_i];
                    float b = s_iv[s][b_i];
                    h0[s * KN + k] = c.x + c.y * a + c.z * b + c.w * (a * b);
                }
            }
        }
        __syncthreads();

        // ---------------- layers 1..3 (gather from previous layer) --------
        int cur = 0;
        for (int l = 1; l < NLAYERS; ++l) {
            const float*  hin  = smem + cur * (S * KN);
            float*        hout = smem + (1 - cur) * (S * KN);
            const int*    pa   = ia + (size_t)(l - 1) * KN;
            const int*    pb   = ib + (size_t)(l - 1) * KN;
            const float4* pc   = coef + (size_t)l * KN;
            for (int k = tid; k < KN; k += BLOCK) {
                int    a_i = pa[k];
                int    b_i = pb[k];
                float4 c   = pc[k];
                #pragma unroll
                for (int s = 0; s < S; ++s) {
                    float a = hin[s * KN + a_i];
                    float b = hin[s * KN + b_i];
                    hout[s * KN + k] = c.x + c.y * a + c.z * b + c.w * (a * b);
                }
            }
            cur = 1 - cur;
            __syncthreads();
        }

        // ---------------- class-sum reduction via WMMA --------------------
        // Row m of the 16x16 accumulator covers segment:
        //   s = m>>2, class = (m>>1)&1, half = m&1  -> 1024 values each.
        // B = ones(4x16)  =>  D[m][n] = sum_k A[m][k]  (+C), all columns equal.
        // Wave w accumulates chunk iterations [w*32, w*32+32).
        {
            const float* h = smem + cur * (S * KN);
            int m    = lane & 15;
            int s    = m >> 2;
            int c    = (m >> 1) & 1;
            int half = m & 1;
            const float* base = h + s * KN + c * 2048 + half * 1024
                                  + ((lane >= 16) ? 2 : 0);

            v2f ones; ones[0] = 1.0f; ones[1] = 1.0f;
            v8f acc0 = {};
            v8f acc1 = {};
            #pragma unroll 4
            for (int j = 0; j < 32; j += 2) {
                int it0 = (wave * 32 + j) * 4;
                v2f a0, a1;
                a0[0] = base[it0 + 0];
                a0[1] = base[it0 + 1];
                a1[0] = base[it0 + 4];
                a1[1] = base[it0 + 5];
                acc0 = __builtin_amdgcn_wmma_f32_16x16x4_f32(
                    false, a0, false, ones, (short)0, acc0, false, false);
                acc1 = __builtin_amdgcn_wmma_f32_16x16x4_f32(
                    false, a1, false, ones, (short)0, acc1, false, false);
            }
            v8f acc = acc0 + acc1;
            // Column 0 of D: lane 0 holds rows 0..7, lane 16 holds rows 8..15.
            if ((lane & 15) == 0) {
                int rbase = (lane >> 4) * 8;
                #pragma unroll
                for (int r = 0; r < 8; ++r) s_part[wave][rbase + r] = acc[r];
            }
        }
        __syncthreads();

        if (pass == 0) {
            // rt0, then the reference's threshold/update step
            if (tid < S * 2) {
                int s = tid >> 1, c = tid & 1;
                float r = 0.f;
                #pragma unroll
                for (int w = 0; w < 8; ++w)
                    r += s_part[w][4 * s + 2 * c] + s_part[w][4 * s + 2 * c + 1];
                s_rt[s][c] = r;   // TAU == 1.0
            }
            __syncthreads();
            if (tid < S * 4) {
                int s = tid >> 2, i = tid & 3;
                float r0 = s_rt[s][0], r1 = s_rt[s][1];
                float thresh = (r1 > r0) ? 0.5f : -0.5f;   // argmax, first-max ties
                float ivv = s_iv[s][i];
                bool  upd = (ivv == r0) || (ivv == r1);    // exact fp equality, as ref
                float tx  = (s_mx[s][i] > thresh) ? 1.0f : 0.0f;
                s_iv[s][i] = upd ? tx : ivv;
            }
            __syncthreads();
        } else {
            // rt1 -> output
            if (tid < S * 2) {
                int s = tid >> 1, c = tid & 1;
                float r = 0.f;
                #pragma unroll
                for (int w = 0; w < 8; ++w)
                    r += s_part[w][4 * s + 2 * c] + s_part[w][4 * s + 2 * c + 1];
                out[(size_t)(sbase + s) * 2 + c] = r;
            }
        }
    }
}

// ---------------------------------------------------------------------------
extern "C" void kernel_launch(void* const* d_in, const int* in_sizes, int n_in,
                              void* d_out, int out_size, void* d_ws, size_t ws_size,
                              hipStream_t stream) {
    (void)in_sizes; (void)n_in; (void)out_size; (void)ws_size;
    const float* x   = (const float*)d_in[0];
    const float* w0  = (const float*)d_in[1];
    const float* wsp = (const float*)d_in[2];
    const int*   ia0 = (const int*)d_in[3];
    const int*   ib0 = (const int*)d_in[4];
    const int*   ia  = (const int*)d_in[5];
    const int*   ib  = (const int*)d_in[6];
    float*       out = (float*)d_out;
    float4*      coef = (float4*)d_ws;   // 4*4096*16B = 256 KB scratch

    difflogic_coeff_kernel<<<(NLAYERS * KN + 255) / 256, 256, 0, stream>>>(
        w0, wsp, coef);

    size_t shmem = (size_t)2 * S * KN * sizeof(float);   // 128 KB dynamic LDS
    (void)hipFuncSetAttribute((const void*)difflogic_model_kernel,
                              hipFuncAttributeMaxDynamicSharedMemorySize,
                              (int)shmem);
    difflogic_model_kernel<<<NS / S, BLOCK, shmem, stream>>>(
        x, coef, ia0, ib0, ia, ib, out);
}